// GCN_28140625723733
// MI455X (gfx1250) — compile-verified
//
#include <hip/hip_runtime.h>
#include <hip/hip_bf16.h>

typedef __attribute__((ext_vector_type(2))) float v2f;
typedef __attribute__((ext_vector_type(8))) float v8f;

#define N_NODES 100000
#define LN_EPS 1e-5f

// ---------------------------------------------------------------------------
// Degree / norm kernels
// ---------------------------------------------------------------------------
__global__ __launch_bounds__(256) void gcn_init_deg(float* __restrict__ deg, int n) {
    int i = blockIdx.x * blockDim.x + threadIdx.x;
    if (i < n) deg[i] = 1.0f;  // self loop contributes 1 to every node's degree
}

__global__ __launch_bounds__(256) void gcn_count_deg(const int* __restrict__ dst,
                                                     float* __restrict__ deg, int nE) {
    int i = blockIdx.x * blockDim.x + threadIdx.x;
    if (i < nE) atomicAdd(&deg[dst[i]], 1.0f);
}

__global__ __launch_bounds__(256) void gcn_dinv(const float* __restrict__ deg,
                                                float* __restrict__ dinv, int n) {
    int i = blockIdx.x * blockDim.x + threadIdx.x;
    if (i < n) {
        float d = deg[i];
        dinv[i] = (d > 0.0f) ? rsqrtf(d) : 0.0f;
    }
}

// ---------------------------------------------------------------------------
// GEMM: H[nrows x OUT_C] = X[nrows x IN_C] @ W[IN_C x OUT_C]  (fp32 WMMA)
// One wave computes a 16-row panel across all OUT_C columns.
// A fragment (16x4 f32): lanes 0-15 -> K={k,k+1}, lanes 16-31 -> K={k+2,k+3}
// B fragment (4x16 f32): same K split; N = lane & 15
// C/D (16x16 f32): 8 VGPRs, row = r + 8*(lane>>4), col = lane & 15
//
// W is staged in LDS with K-pairs interleaved:
//   Wt[((k>>1)*OUT_C + n)*2 + (k&1)] = W[k][n]
// so each lane's B fragment {W[k0][n], W[k0+1][n]} is ONE contiguous,
// 8B-aligned ds_load_b64 into an even-aligned VGPR pair (no repack movs).
// ---------------------------------------------------------------------------
template <int IN_C, int OUT_C>
__global__ __launch_bounds__(256) void gcn_gemm_wmma(const float* __restrict__ X,
                                                     const float* __restrict__ W,
                                                     float* __restrict__ H, int nrows) {
    constexpr int NT = OUT_C / 16;
    __shared__ float Wlds[IN_C * OUT_C];

    // Stage W into LDS in K-pair-interleaved layout
    for (int i = threadIdx.x; i < IN_C * OUT_C; i += 256) {
        const int kk = i / OUT_C;
        const int n  = i % OUT_C;
        Wlds[((kk >> 1) * OUT_C + n) * 2 + (kk & 1)] = W[i];
    }
    __syncthreads();

    const int wave  = threadIdx.x >> 5;
    const int lane  = threadIdx.x & 31;
    const int half  = lane >> 4;   // 0: K pair {0,1}; 1: K pair {2,3}
    const int lrow  = lane & 15;
    const int mtile = blockIdx.x * 8 + wave;
    if (mtile * 16 >= nrows) return;

    const float* __restrict__ xrow = X + (size_t)(mtile * 16 + lrow) * IN_C;

    v8f zero = {};
    v8f acc[NT];
#pragma unroll
    for (int t = 0; t < NT; ++t) acc[t] = zero;

    for (int k = 0; k < IN_C; k += 4) {
        const int k0 = k + 2 * half;
        v2f a;
        a.x = xrow[k0];
        a.y = xrow[k0 + 1];
        const int kp = k0 >> 1;  // pair index
#pragma unroll
        for (int t = 0; t < NT; ++t) {
            const v2f b = *(const v2f*)&Wlds[(kp * OUT_C + t * 16 + lrow) * 2];
            acc[t] = __builtin_amdgcn_wmma_f32_16x16x4_f32(
                false, a, false, b, (short)0, acc[t], false, false);
        }
    }

#pragma unroll
    for (int t = 0; t < NT; ++t) {
#pragma unroll
        for (int r = 0; r < 8; ++r) {
            const int m = mtile * 16 + r + 8 * half;
            H[(size_t)m * OUT_C + t * 16 + lrow] = acc[t][r];
        }
    }
}

// ---------------------------------------------------------------------------
// OUT[i][f] = bias[f] + dinv[i]^2 * H[i][f]   (bias + self-loop term)
// ---------------------------------------------------------------------------
template <int C>
__global__ __launch_bounds__(256) void gcn_init_out(const float* __restrict__ H,
                                                    const float* __restrict__ bias,
                                                    const float* __restrict__ dinv,
                                                    float* __restrict__ OUT, int n) {
    long long idx = (long long)blockIdx.x * blockDim.x + threadIdx.x;
    if (idx >= (long long)n * C) return;
    int node = (int)(idx / C);
    int f    = (int)(idx % C);
    float di = dinv[node];
    OUT[idx] = bias[f] + di * di * H[idx];
}

// ---------------------------------------------------------------------------
// Edge scatter: OUT[dst] += dinv[src]*dinv[dst] * H[src]   (one wave per edge)
// ---------------------------------------------------------------------------
template <int C>
__global__ __launch_bounds__(256) void gcn_scatter(const float* __restrict__ H,
                                                   const int* __restrict__ src,
                                                   const int* __restrict__ dst,
                                                   const float* __restrict__ dinv,
                                                   float* __restrict__ OUT, int nE) {
    constexpr int FPL = C / 32;  // floats per lane
    const int gwave = (int)(((long long)blockIdx.x * blockDim.x + threadIdx.x) >> 5);
    const int lane  = threadIdx.x & 31;
    if (gwave >= nE) return;
    const int s = src[gwave];
    const int d = dst[gwave];
    const float norm = dinv[s] * dinv[d];
    const float* __restrict__ h = H + (size_t)s * C + lane * FPL;
    float* __restrict__ o       = OUT + (size_t)d * C + lane * FPL;
#pragma unroll
    for (int i = 0; i < FPL; ++i) atomicAdd(o + i, norm * h[i]);
}

// ---------------------------------------------------------------------------
// In-place ReLU + LayerNorm over 128 features; one wave per node (float4/lane)
// ---------------------------------------------------------------------------
__global__ __launch_bounds__(256) void gcn_relu_ln(float* __restrict__ X,
                                                   const float* __restrict__ gamma,
                                                   const float* __restrict__ beta, int n) {
    const int gwave = (int)(((long long)blockIdx.x * blockDim.x + threadIdx.x) >> 5);
    const int lane  = threadIdx.x & 31;
    if (gwave >= n) return;
    float* row = X + (size_t)gwave * 128;
    float4 v = ((float4*)row)[lane];
    v.x = fmaxf(v.x, 0.0f);
    v.y = fmaxf(v.y, 0.0f);
    v.z = fmaxf(v.z, 0.0f);
    v.w = fmaxf(v.w, 0.0f);
    float s1 = v.x + v.y + v.z + v.w;
    float s2 = v.x * v.x + v.y * v.y + v.z * v.z + v.w * v.w;
#pragma unroll
    for (int m = 16; m >= 1; m >>= 1) {
        s1 += __shfl_xor(s1, m, 32);
        s2 += __shfl_xor(s2, m, 32);
    }
    const float mu  = s1 * (1.0f / 128.0f);
    const float var = s2 * (1.0f / 128.0f) - mu * mu;
    const float r   = rsqrtf(var + LN_EPS);
    const float4 g  = ((const float4*)gamma)[lane];
    const float4 be = ((const float4*)beta)[lane];
    v.x = (v.x - mu) * r * g.x + be.x;
    v.y = (v.y - mu) * r * g.y + be.y;
    v.z = (v.z - mu) * r * g.z + be.z;
    v.w = (v.w - mu) * r * g.w + be.w;
    ((float4*)row)[lane] = v;
}

// ---------------------------------------------------------------------------
extern "C" void kernel_launch(void* const* d_in, const int* in_sizes, int n_in,
                              void* d_out, int out_size, void* d_ws, size_t ws_size,
                              hipStream_t stream) {
    const float* x     = (const float*)d_in[0];
    const int*   edge  = (const int*)d_in[1];
    const float* W1    = (const float*)d_in[2];
    const float* b1    = (const float*)d_in[3];
    const float* gamma = (const float*)d_in[4];
    const float* beta  = (const float*)d_in[5];
    const float* W2    = (const float*)d_in[6];
    const float* b2    = (const float*)d_in[7];
    float* out = (float*)d_out;

    const int N = N_NODES;
    const int E = in_sizes[1] / 2;
    const int* src = edge;
    const int* dst = edge + E;

    // Workspace layout (fp32): deg[N] | dinv[N] | H1[N*128] | OUT1[N*128]
    float* ws   = (float*)d_ws;
    float* deg  = ws;
    float* dinv = deg + N;
    float* H1   = dinv + N;                 // layer-1 GEMM out; reused as layer-2 GEMM out
    float* OUT1 = H1 + (size_t)N * 128;     // layer-1 aggregated; LN'd in-place

    // 1) degrees + normalization
    gcn_init_deg<<<(N + 255) / 256, 256, 0, stream>>>(deg, N);
    gcn_count_deg<<<(E + 255) / 256, 256, 0, stream>>>(dst, deg, E);
    gcn_dinv<<<(N + 255) / 256, 256, 0, stream>>>(deg, dinv, N);

    // 2) H1 = x @ W1   (WMMA f32)
    const int mtiles = (N + 15) / 16;
    gcn_gemm_wmma<128, 128><<<(mtiles + 7) / 8, 256, 0, stream>>>(x, W1, H1, N);

    // 3) OUT1 = b1 + dinv^2 * H1 ; then scatter edges
    {
        long long tot = (long long)N * 128;
        gcn_init_out<128><<<(int)((tot + 255) / 256), 256, 0, stream>>>(H1, b1, dinv, OUT1, N);
        long long etot = (long long)E * 32;
        gcn_scatter<128><<<(int)((etot + 255) / 256), 256, 0, stream>>>(H1, src, dst, dinv, OUT1, E);
    }

    // 4) ReLU + LayerNorm in place
    {
        long long tot = (long long)N * 32;
        gcn_relu_ln<<<(int)((tot + 255) / 256), 256, 0, stream>>>(OUT1, gamma, beta, N);
    }

    // 5) H3 = OUT1 @ W2  (reuse H1 buffer)
    gcn_gemm_wmma<128, 64><<<(mtiles + 7) / 8, 256, 0, stream>>>(OUT1, W2, H1, N);

    // 6) out = b2 + dinv^2 * H3 ; then scatter edges
    {
        long long tot = (long long)N * 64;
        gcn_init_out<64><<<(int)((tot + 255) / 256), 256, 0, stream>>>(H1, b2, dinv, out, N);
        long long etot = (long long)E * 32;
        gcn_scatter<64><<<(int)((etot + 255) / 256), 256, 0, stream>>>(H1, src, dst, dinv, out, E);
    }
}